// GraphConvolution_sparsemax_exp_27066883900218
// MI455X (gfx1250) — compile-verified
//
#include <hip/hip_runtime.h>
#include <hip/hip_bf16.h>

// ---------------------------------------------------------------------------
// GraphConvolution + sparsemax coefficient combine, fused for MI455X (gfx1250)
//
// out = theta*(support @ W) + (1-theta)*support
// support = (1-alpha) * (sum_k cf_k * A_k) @ X + alpha * H0
// cf = sparsemax(exp(pg_coeffs)), theta = log(lamda/l + 1)
//
// Roofline: reading adj_powers (10*4096^2*4B = 671 MB) dominates ->
// ~29 us @ 23.3 TB/s HBM.  GEMM math is only ~4.4 GFLOP fp32, far below any
// compute roof, so fp32 WMMA 16x16x4 keeps full precision at zero cost.
// Single pass over A; A stream marked non-temporal so X/W stay L2-resident.
// ---------------------------------------------------------------------------

typedef float v2f __attribute__((ext_vector_type(2)));
typedef float v4f __attribute__((ext_vector_type(4)));
typedef float v8f __attribute__((ext_vector_type(8)));

#define NN   4096   // num nodes
#define DD   128    // in_features == out_features
#define KK   10     // gpr_coeff
#define BM   32     // rows per block
#define KC   64     // j-chunk per LDS stage

// LDS strides (floats), padded for bank-conflict-free WMMA fragment loads:
//  A tile  [32 x 64]  stride 68  (4-bank shift/row; b64 A-frags hit 32 banks)
//  X tile  [64 x 128] stride 136 (8-bank shift/row; b32 B-frags hit 32 banks)
//  S tile  [32 x 128] stride 132 (4-bank shift/row, rows 8B-aligned)
#define AST  68
#define XST  136
#define SST  132
#define SMEM_FLOATS (BM*AST + KC*XST)   // 2176 + 8704 = 10880 floats = 43520 B

// ---------------------------------------------------------------------------
// Kernel 1: sparsemax(exp(pg)) on K=10 values + scalar prep.  One wave.
// ws[0..9] = (1-alpha)*cf_k,  ws[10] = alpha,  ws[11] = theta
// ---------------------------------------------------------------------------
__global__ void gcn_prep_kernel(const float* __restrict__ pg,
                                const float* __restrict__ lamda_p,
                                const float* __restrict__ alpha_p,
                                const int*   __restrict__ l_p,
                                float* __restrict__ ws) {
    if (threadIdx.x != 0 || blockIdx.x != 0) return;
    float z[KK], zs[KK];
    #pragma unroll
    for (int k = 0; k < KK; ++k) { z[k] = expf(pg[k]); zs[k] = z[k]; }
    // descending insertion sort (K=10, trivial)
    for (int i = 1; i < KK; ++i) {
        float key = zs[i]; int j = i - 1;
        while (j >= 0 && zs[j] < key) { zs[j + 1] = zs[j]; --j; }
        zs[j + 1] = key;
    }
    // sparsemax support / tau
    float cum = 0.0f, css_at = zs[0];
    int ksup = 1;
    for (int k = 1; k <= KK; ++k) {
        cum += zs[k - 1];
        if (1.0f + (float)k * zs[k - 1] > cum) { ksup = k; css_at = cum; }
    }
    float tau = (css_at - 1.0f) / (float)ksup;
    float alpha = alpha_p[0];
    float theta = logf(lamda_p[0] / (float)l_p[0] + 1.0f);
    #pragma unroll
    for (int k = 0; k < KK; ++k) {
        float cf = z[k] - tau;
        ws[k] = (1.0f - alpha) * (cf > 0.0f ? cf : 0.0f);
    }
    ws[10] = alpha;
    ws[11] = theta;
}

// ---------------------------------------------------------------------------
// Kernel 2: fully fused combine + spmm + residual + dense GEMM + blend.
// Grid: NN/BM = 128 blocks, 256 threads (8 wave32) each.
// Wave w owns ALL 32 rows x 16 cols (n0 = w*16): two accumulators split
// along M (rows 0-15 and 16-31) SHARING one B fragment per k-step, so the
// B fragment (.x = row k, .y = row k+1, same column) is a single
// ds_load_2addr_b32 with no register shuffles, and each A fragment is one
// b64 load.
// WMMA f32 16x16x4 fragment mapping (wave32, lane = 16*p + q):
//   A: VGPR0 = A[m=q][k=2p],   VGPR1 = A[m=q][k=2p+1]     (v2f, one b64 load)
//   B: VGPR0 = B[k=2p][n=q],   VGPR1 = B[k=2p+1][n=q]
//   C/D: VGPR g = C[m=g+8p][n=q]
// ---------------------------------------------------------------------------
__global__ __launch_bounds__(256, 2)
void gcn_fused_kernel(const float* __restrict__ X,    // [NN, DD]
                      const float* __restrict__ H0,   // [NN, DD]
                      const float* __restrict__ A,    // [KK, NN, NN]
                      const float* __restrict__ W,    // [DD, DD]
                      const float* __restrict__ cfws, // ws scalars
                      float* __restrict__ out) {      // [NN, DD]
    __shared__ float smem[SMEM_FLOATS];
    float* As = smem;              // [BM][AST]
    float* Xs = smem + BM * AST;   // [KC][XST]

    const int tid  = threadIdx.x;
    const int i0   = blockIdx.x * BM;
    const int wave = tid >> 5;
    const int lane = tid & 31;
    const int p    = lane >> 4;      // half-wave
    const int q    = lane & 15;      // m (for A) / n (for B,C,D)
    const int n0   = wave * 16;      // this wave's 16-column strip

    float cfw[KK];
    #pragma unroll
    for (int k = 0; k < KK; ++k) cfw[k] = cfws[k];
    const float alpha = cfws[10];
    const float theta = cfws[11];

    v8f accA = {};   // rows 0-15  of the block tile
    v8f accB = {};   // rows 16-31 of the block tile

    for (int jc = 0; jc < NN; jc += KC) {
        // ---- stage combined A tile: As[r][c] = sum_k cfw[k]*A_k[i0+r][jc+c]
        //      A is a stream-once 671 MB read: non-temporal so X/W stay in L2.
        #pragma unroll
        for (int it = 0; it < 2; ++it) {
            int c   = tid + it * 256;          // 512 float4 chunks
            int row = c >> 4;
            int c4  = (c & 15) << 2;
            const float* gp = A + (size_t)(i0 + row) * NN + jc + c4;
            v4f v = {0.f, 0.f, 0.f, 0.f};
            #pragma unroll
            for (int k = 0; k < KK; ++k) {
                v4f a4 = __builtin_nontemporal_load(
                             (const v4f*)(gp + (size_t)k * NN * NN));
                v += cfw[k] * a4;
            }
            *(v4f*)(As + row * AST + c4) = v;
        }
        // ---- stage X tile (L2-resident, regular temporal)
        #pragma unroll
        for (int it = 0; it < 8; ++it) {
            int c   = tid + it * 256;          // 2048 float4 chunks
            int row = c >> 5;
            int c4  = (c & 31) << 2;
            *(v4f*)(Xs + row * XST + c4) =
                *(const v4f*)(X + (size_t)(jc + row) * DD + c4);
        }
        __syncthreads();

        // ---- WMMA f32 16x16x4 chain over this chunk
        #pragma unroll
        for (int js = 0; js < KC; js += 4) {
            v2f aA = *(const v2f*)(As + q        * AST + js + 2 * p);
            v2f aB = *(const v2f*)(As + (16 + q) * AST + js + 2 * p);
            v2f b;
            b.x = Xs[(js + 2 * p)     * XST + n0 + q];
            b.y = Xs[(js + 2 * p + 1) * XST + n0 + q];
            accA = __builtin_amdgcn_wmma_f32_16x16x4_f32(
                       false, aA, false, b, (short)0, accA, false, false);
            accB = __builtin_amdgcn_wmma_f32_16x16x4_f32(
                       false, aB, false, b, (short)0, accB, false, false);
        }
        __syncthreads();
    }

    // ---- epilogue: support = acc + alpha*H0 -> LDS (reuse smem as S[32][SST])
    float* Ss = smem;
    #pragma unroll
    for (int g = 0; g < 8; ++g) {
        int ra = g + 8 * p;          // local row for accA (0..15)
        int rb = 16 + ra;            // local row for accB (16..31)
        float h0a = H0[(size_t)(i0 + ra) * DD + n0 + q];
        float h0b = H0[(size_t)(i0 + rb) * DD + n0 + q];
        Ss[ra * SST + n0 + q] = accA[g] + alpha * h0a;
        Ss[rb * SST + n0 + q] = accB[g] + alpha * h0b;
    }
    __syncthreads();

    // ---- small GEMM: o = support_tile @ W  (W streamed from hot L2)
    v8f oA = {};
    v8f oB = {};
    #pragma unroll
    for (int ks = 0; ks < DD; ks += 4) {
        v2f aA = *(const v2f*)(Ss + q        * SST + ks + 2 * p);
        v2f aB = *(const v2f*)(Ss + (16 + q) * SST + ks + 2 * p);
        const float* Wr = W + (size_t)(ks + 2 * p) * DD + n0 + q;
        v2f b;
        b.x = Wr[0];
        b.y = Wr[DD];
        oA = __builtin_amdgcn_wmma_f32_16x16x4_f32(
                 false, aA, false, b, (short)0, oA, false, false);
        oB = __builtin_amdgcn_wmma_f32_16x16x4_f32(
                 false, aB, false, b, (short)0, oB, false, false);
    }

    // ---- blend + store: out = theta*o + (1-theta)*support
    const float omt = 1.0f - theta;
    #pragma unroll
    for (int g = 0; g < 8; ++g) {
        int ra = g + 8 * p;
        int rb = 16 + ra;
        float sA = Ss[ra * SST + n0 + q];
        float sB = Ss[rb * SST + n0 + q];
        out[(size_t)(i0 + ra) * DD + n0 + q] = theta * oA[g] + omt * sA;
        out[(size_t)(i0 + rb) * DD + n0 + q] = theta * oB[g] + omt * sB;
    }
}

// ---------------------------------------------------------------------------
extern "C" void kernel_launch(void* const* d_in, const int* in_sizes, int n_in,
                              void* d_out, int out_size, void* d_ws, size_t ws_size,
                              hipStream_t stream) {
    const float* X   = (const float*)d_in[0];  // input   [NN,DD]
    const float* H0  = (const float*)d_in[1];  // h0      [NN,DD]
    const float* A   = (const float*)d_in[2];  // adj_pow [KK,NN,NN]
    const float* PG  = (const float*)d_in[3];  // pg_coeffs [KK,1]
    const float* W   = (const float*)d_in[4];  // weight  [DD,DD]
    const float* LAM = (const float*)d_in[5];  // lamda (scalar)
    const float* ALP = (const float*)d_in[6];  // alpha (scalar)
    const int*   L   = (const int*)d_in[7];    // l (scalar int)
    float* ws  = (float*)d_ws;
    float* out = (float*)d_out;

    gcn_prep_kernel<<<1, 32, 0, stream>>>(PG, LAM, ALP, L, ws);
    gcn_fused_kernel<<<NN / BM, 256, 0, stream>>>(X, H0, A, W, ws, out);
}